// RegionProposalNetwork_75711683493960
// MI455X (gfx1250) — compile-verified
//
#include <hip/hip_runtime.h>
#include <hip/hip_bf16.h>
#include <math.h>

typedef __attribute__((ext_vector_type(2))) float v2f;
typedef __attribute__((ext_vector_type(8))) float v8f;

#define BATCH     8
#define CIN       256
#define COUT      256
#define FH        64
#define FW        64
#define IMG_W     1024.0f
#define IMG_H     1024.0f
#define NA        9
#define NANCH     (FH*FW*NA)          /* 36864 */
#define NPAD      65536
#define PRE_NMSK  2000
#define POST_NMSK 1000
#define NMS_THRV  0.7f
#define MIN_SIZEV 0.001f
#define BBOX_CLIP 4.135166556742356f  /* log(1000/16) */
#define STRIDEF   16.0f

// ---------------------------------------------------------------------------
// Kernel 0: one-time weight reorder  w1[cout][cin][tap] -> wT[chunk][tap][ck][cout]
// so that each conv K-step stages a fully contiguous 36 KB block.
// ---------------------------------------------------------------------------
__global__ void rpn_w1_transpose(const float* __restrict__ w1,
                                 float* __restrict__ wT)
{
  const int t = blockIdx.x * blockDim.x + threadIdx.x;
  if (t >= 2304 * 256) return;
  const int chunk = t / 9216;           // cin chunk (4 channels)
  const int rem   = t % 9216;
  const int tap   = rem >> 10;
  const int rem2  = rem & 1023;
  const int ck    = rem2 >> 8;
  const int co    = rem2 & 255;
  wT[t] = w1[(size_t)co*2304 + (size_t)(chunk*4 + ck)*9 + tap];
}

// ---------------------------------------------------------------------------
// Kernel 1: 3x3 conv (256->256) + bias + ReLU as implicit GEMM on fp32 WMMA.
// One block per (batch, y-row). 8 waves; wave w owns couts [32w, 32w+32).
// Each wave: 4 M-tiles (x = 0..63 in 16-chunks) x 2 N-tiles -> 8 v8f accs.
// Weights staged per K-step via global_load_async_to_lds_b128 (ASYNCcnt).
// ---------------------------------------------------------------------------
__global__ __launch_bounds__(256) void rpn_conv3x3_wmma(
    const float* __restrict__ feat, const float* __restrict__ wT,
    const float* __restrict__ b1, float* __restrict__ tbuf)
{
  __shared__ float sIn[4*3*66];                                 // [ck][ry][cx]
  __shared__ float sW[9*4*256] __attribute__((aligned(16)));    // [tap][ck][cout]

  const int b    = blockIdx.x / FH;
  const int y    = blockIdx.x % FH;
  const int tid  = threadIdx.x;
  const int lane = tid & 31;
  const int wv   = tid >> 5;
  const int nlane = lane & 15;
  const int kb    = (lane < 16) ? 0 : 2;  // A/B K-half per ISA 16x4 fp32 layout
  const int coutBase = wv * 32;

  const v8f vzero = {};
  v8f acc[4][2];
#pragma unroll
  for (int mt = 0; mt < 4; ++mt)
#pragma unroll
    for (int nt = 0; nt < 2; ++nt) acc[mt][nt] = vzero;

  for (int c0 = 0; c0 < CIN; c0 += 4) {
    __syncthreads();  // previous chunk fully consumed before overwriting LDS

    // Kick off async weight DMA: contiguous 36 KB chunk, 16 B per lane per issue.
    {
      const float* wsrc = wT + (size_t)(c0 >> 2) * 9216;
      for (int e4 = tid * 4; e4 < 9216; e4 += 1024) {
        const unsigned ldsa = (unsigned)(unsigned long long)&sW[e4];
        const unsigned voff = (unsigned)(e4 * 4);
        asm volatile("global_load_async_to_lds_b128 %0, %1, %2 offset:0"
                     :: "v"(ldsa), "v"(voff), "s"(wsrc)
                     : "memory");
      }
    }

    // Stage 4-channel input slab with halo (zero padded) on the VALU path.
    for (int e = tid; e < 4*3*66; e += 256) {
      const int ck = e / 198; const int rem = e % 198;
      const int ry = rem / 66; const int cx = rem % 66;
      const int yy = y - 1 + ry; const int xx = cx - 1;
      float v = 0.f;
      if (yy >= 0 && yy < FH && xx >= 0 && xx < FW)
        v = feat[(((size_t)b*CIN + (c0 + ck))*FH + yy)*FW + xx];
      sIn[e] = v;
    }

    asm volatile("s_wait_asynccnt 0x0" ::: "memory");
    __syncthreads();

#pragma unroll
    for (int tap = 0; tap < 9; ++tap) {
      const int dy = tap / 3, dxx = tap % 3;
      const float* wp = &sW[tap * 1024];
      v2f bf0, bf1;
      bf0[0] = wp[(kb + 0)*256 + coutBase + nlane];
      bf0[1] = wp[(kb + 1)*256 + coutBase + nlane];
      bf1[0] = wp[(kb + 0)*256 + coutBase + 16 + nlane];
      bf1[1] = wp[(kb + 1)*256 + coutBase + 16 + nlane];
#pragma unroll
      for (int mt = 0; mt < 4; ++mt) {
        const int col = mt*16 + nlane + dxx;   // sIn col: (x + dxx - 1) + 1
        v2f af;
        af[0] = sIn[(kb + 0)*198 + dy*66 + col];
        af[1] = sIn[(kb + 1)*198 + dy*66 + col];
        acc[mt][0] = __builtin_amdgcn_wmma_f32_16x16x4_f32(
            false, af, false, bf0, (short)0, acc[mt][0], false, false);
        acc[mt][1] = __builtin_amdgcn_wmma_f32_16x16x4_f32(
            false, af, false, bf1, (short)0, acc[mt][1], false, false);
      }
    }
  }

  // Epilogue: bias + ReLU, store NCHW. C/D map: vgpr r -> M=r (lanes<16) / r+8.
#pragma unroll
  for (int nt = 0; nt < 2; ++nt) {
    const int co = coutBase + nt*16 + nlane;
    const float bias = b1[co];
    float* out = &tbuf[(((size_t)b*COUT + co)*FH + y)*FW];
    const int xoff = (lane < 16) ? 0 : 8;
#pragma unroll
    for (int mt = 0; mt < 4; ++mt)
#pragma unroll
      for (int r = 0; r < 8; ++r)
        out[mt*16 + xoff + r] = fmaxf(acc[mt][nt][r] + bias, 0.f);
  }
}

// ---------------------------------------------------------------------------
// Kernel 2: fused 1x1 heads (obj 9 + box 36) + sigmoid + anchor decode.
// One block per (batch, 4 rows); one thread per (row, x). Weights in LDS.
// ---------------------------------------------------------------------------
__global__ __launch_bounds__(256) void rpn_head_decode(
    const float* __restrict__ tbuf,
    const float* __restrict__ w_obj, const float* __restrict__ b_obj,
    const float* __restrict__ w_box, const float* __restrict__ b_box,
    float* __restrict__ scores, float* __restrict__ boxes)
{
  __shared__ float sW[45*256];
  __shared__ float sB[45];
  const int b = blockIdx.x / (FH/4);
  const int y = (blockIdx.x % (FH/4)) * 4 + (threadIdx.x >> 6);
  const int x = threadIdx.x & 63;

  for (int e = threadIdx.x; e < 45*256; e += 256) {
    const int o = e >> 8, c = e & 255;
    sW[e] = (o < 9) ? w_obj[o*256 + c] : w_box[(o - 9)*256 + c];
  }
  if (threadIdx.x < 45)
    sB[threadIdx.x] = (threadIdx.x < 9) ? b_obj[threadIdx.x] : b_box[threadIdx.x - 9];
  __syncthreads();

  float acc[45];
#pragma unroll
  for (int o = 0; o < 45; ++o) acc[o] = 0.f;
  const float* tp = &tbuf[(((size_t)b*COUT)*FH + y)*FW + x];
  for (int c = 0; c < 256; ++c) {
    const float v = tp[(size_t)c * FH * FW];
#pragma unroll
    for (int o = 0; o < 45; ++o) acc[o] += v * sW[o*256 + c];
  }

#pragma unroll
  for (int a = 0; a < 9; ++a) {
    const float logit = acc[a] + sB[a];
    const float score = 1.f / (1.f + expf(-logit));
    const float ddx = acc[9 + a*4 + 0] + sB[9 + a*4 + 0];
    const float ddy = acc[9 + a*4 + 1] + sB[9 + a*4 + 1];
    float dw = acc[9 + a*4 + 2] + sB[9 + a*4 + 2];
    float dh = acc[9 + a*4 + 3] + sB[9 + a*4 + 3];
    const int ridx = a / 3, sidx = a % 3;
    const float ratio = (ridx == 0) ? 0.5f : ((ridx == 1) ? 1.0f : 2.0f);
    const float scale = (sidx == 0) ? 128.f : ((sidx == 1) ? 256.f : 512.f);
    const float hr = sqrtf(ratio);
    const float aw = (1.f / hr) * scale;
    const float ah = hr * scale;
    const float acx = (float)x * STRIDEF;
    const float acy = (float)y * STRIDEF;
    dw = fminf(dw, BBOX_CLIP);
    dh = fminf(dh, BBOX_CLIP);
    const float cx = ddx * aw + acx;
    const float cy = ddy * ah + acy;
    const float ww = expf(dw) * aw;
    const float hh = expf(dh) * ah;
    const float x1 = fminf(fmaxf(cx - 0.5f*ww, 0.f), IMG_W);
    const float y1 = fminf(fmaxf(cy - 0.5f*hh, 0.f), IMG_H);
    const float x2 = fminf(fmaxf(cx + 0.5f*ww, 0.f), IMG_W);
    const float y2 = fminf(fmaxf(cy + 0.5f*hh, 0.f), IMG_H);
    const bool ok = ((x2 - x1) > MIN_SIZEV) && ((y2 - y1) > MIN_SIZEV);
    const int n = (y*FW + x)*NA + a;
    scores[(size_t)b*NANCH + n] = ok ? score : -1.0f;
    float* bp = &boxes[((size_t)b*NANCH + n)*4];
    bp[0] = x1; bp[1] = y1; bp[2] = x2; bp[3] = y2;
  }
}

// ---------------------------------------------------------------------------
// Top-k via 64-bit key bitonic sort: hi = monotone float flip of score,
// lo = ~index (descending key sort == descending score, stable index ties).
// ---------------------------------------------------------------------------
__global__ void rpn_keys(const float* __restrict__ scores,
                         unsigned long long* __restrict__ keys)
{
  const int t = blockIdx.x * blockDim.x + threadIdx.x;
  if (t >= BATCH * NPAD) return;
  const int b = t >> 16;
  const int r = t & (NPAD - 1);
  unsigned long long key = 0ull;
  if (r < NANCH) {
    unsigned u = __float_as_uint(scores[(size_t)b*NANCH + r]);
    u = (u & 0x80000000u) ? ~u : (u | 0x80000000u);
    key = ((unsigned long long)u << 32) | (unsigned)(~r);
  }
  keys[t] = key;
}

__global__ void rpn_bitonic(unsigned long long* __restrict__ keys, int k, int j)
{
  const int t = blockIdx.x * blockDim.x + threadIdx.x;
  const int b = t >> 16;
  const int i = t & (NPAD - 1);
  const int ixj = i ^ j;
  if (ixj <= i) return;
  unsigned long long* kb = keys + ((size_t)b << 16);
  const unsigned long long a = kb[i], c = kb[ixj];
  const bool desc = ((i & k) == 0);
  if (desc ? (a < c) : (a > c)) { kb[i] = c; kb[ixj] = a; }
}

__global__ void rpn_gather(const unsigned long long* __restrict__ keys,
                           const float* __restrict__ scores,
                           const float* __restrict__ boxes,
                           float* __restrict__ tboxes, int* __restrict__ keep)
{
  const int t = blockIdx.x * blockDim.x + threadIdx.x;
  if (t >= BATCH * PRE_NMSK) return;
  const int b = t / PRE_NMSK;
  const int r = t % PRE_NMSK;
  const unsigned long long key = keys[((size_t)b << 16) + r];
  const int idx = (int)(~(unsigned)(key & 0xFFFFFFFFull));
  const float sc = scores[(size_t)b*NANCH + idx];
  const float* p = &boxes[((size_t)b*NANCH + idx)*4];
  float* q = &tboxes[((size_t)b*PRE_NMSK + r)*4];
  q[0] = p[0]; q[1] = p[1]; q[2] = p[2]; q[3] = p[3];
  keep[t] = (sc >= 0.f) ? 1 : 0;
}

// ---------------------------------------------------------------------------
// Kernel 5: greedy NMS, one block per batch, boxes + keep bits in LDS.
// ---------------------------------------------------------------------------
__global__ __launch_bounds__(256) void rpn_nms(const float* __restrict__ tboxes,
                                               int* __restrict__ keep)
{
  __shared__ float4 sBox[PRE_NMSK];
  __shared__ int sKeep[PRE_NMSK];
  const int b = blockIdx.x;
  const int tid = threadIdx.x;
  for (int i = tid; i < PRE_NMSK; i += 256) {
    const float* p = &tboxes[((size_t)b*PRE_NMSK + i)*4];
    sBox[i] = make_float4(p[0], p[1], p[2], p[3]);
    sKeep[i] = keep[(size_t)b*PRE_NMSK + i];
  }
  for (int i = 0; i < PRE_NMSK; ++i) {
    __syncthreads();
    if (!sKeep[i]) continue;
    const float4 bi = sBox[i];
    const float areaA = (bi.z - bi.x) * (bi.w - bi.y);
    for (int j = i + 1 + tid; j < PRE_NMSK; j += 256) {
      if (!sKeep[j]) continue;
      const float4 bj = sBox[j];
      const float areaB = (bj.z - bj.x) * (bj.w - bj.y);
      const float ltx = fmaxf(bi.x, bj.x), lty = fmaxf(bi.y, bj.y);
      const float rbx = fminf(bi.z, bj.z), rby = fminf(bi.w, bj.w);
      const float iw = fmaxf(rbx - ltx, 0.f), ih = fmaxf(rby - lty, 0.f);
      const float inter = iw * ih;
      const float uni = fmaxf(areaA + areaB - inter, 1e-9f);
      if (inter / uni > NMS_THRV) sKeep[j] = 0;
    }
  }
  __syncthreads();
  for (int i = tid; i < PRE_NMSK; i += 256)
    keep[(size_t)b*PRE_NMSK + i] = sKeep[i];
}

// ---------------------------------------------------------------------------
// Kernel 6: stable partition kept-first (== stable argsort of !keep), first
// 1000 out; invalid slots zeroed.
// ---------------------------------------------------------------------------
__global__ __launch_bounds__(256) void rpn_output(const float* __restrict__ tboxes,
                                                  const int* __restrict__ keep,
                                                  float* __restrict__ outBoxes,
                                                  float* __restrict__ outValid)
{
  __shared__ int sKeep[PRE_NMSK];
  __shared__ int sPre[PRE_NMSK + 1];
  const int b = blockIdx.x;
  const int tid = threadIdx.x;
  for (int i = tid; i < PRE_NMSK; i += 256)
    sKeep[i] = keep[(size_t)b*PRE_NMSK + i];
  __syncthreads();
  if (tid == 0) {
    int s = 0;
    for (int i = 0; i < PRE_NMSK; ++i) { sPre[i] = s; s += sKeep[i]; }
    sPre[PRE_NMSK] = s;
  }
  __syncthreads();
  const int total = sPre[PRE_NMSK];
  for (int i = tid; i < PRE_NMSK; i += 256) {
    const int dest = sKeep[i] ? sPre[i] : (total + i - sPre[i]);
    if (dest < POST_NMSK) {
      const float* p = &tboxes[((size_t)b*PRE_NMSK + i)*4];
      float* q = &outBoxes[((size_t)b*POST_NMSK + dest)*4];
      const float m = sKeep[i] ? 1.f : 0.f;
      q[0] = p[0]*m; q[1] = p[1]*m; q[2] = p[2]*m; q[3] = p[3]*m;
      outValid[(size_t)b*POST_NMSK + dest] = m;
    }
  }
}

// ---------------------------------------------------------------------------
extern "C" void kernel_launch(void* const* d_in, const int* in_sizes, int n_in,
                              void* d_out, int out_size, void* d_ws, size_t ws_size,
                              hipStream_t stream)
{
  (void)in_sizes; (void)n_in; (void)out_size; (void)ws_size;
  const float* feat  = (const float*)d_in[1];
  const float* w1    = (const float*)d_in[2];
  const float* b1    = (const float*)d_in[3];
  const float* w_obj = (const float*)d_in[4];
  const float* b_obj = (const float*)d_in[5];
  const float* w_box = (const float*)d_in[6];
  const float* b_box = (const float*)d_in[7];

  char* ws = (char*)d_ws;
  size_t off = 0;
  auto carve = [&](size_t bytes) -> void* {
    void* p = ws + off;
    off = (off + bytes + 255) & ~(size_t)255;
    return p;
  };
  float* tbuf   = (float*)carve((size_t)BATCH*COUT*FH*FW*sizeof(float));   // 32 MB
  float* wT     = (float*)carve((size_t)2304*256*sizeof(float));           // 2.36 MB
  float* scores = (float*)carve((size_t)BATCH*NANCH*sizeof(float));
  float* boxes  = (float*)carve((size_t)BATCH*NANCH*4*sizeof(float));
  unsigned long long* keys =
      (unsigned long long*)carve((size_t)BATCH*NPAD*sizeof(unsigned long long));
  float* tboxes = (float*)carve((size_t)BATCH*PRE_NMSK*4*sizeof(float));
  int*   keep   = (int*)carve((size_t)BATCH*PRE_NMSK*sizeof(int));

  rpn_w1_transpose<<<(2304*256)/256, 256, 0, stream>>>(w1, wT);
  rpn_conv3x3_wmma<<<BATCH*FH, 256, 0, stream>>>(feat, wT, b1, tbuf);
  rpn_head_decode<<<BATCH*(FH/4), 256, 0, stream>>>(tbuf, w_obj, b_obj, w_box,
                                                    b_box, scores, boxes);
  rpn_keys<<<(BATCH*NPAD)/256, 256, 0, stream>>>(scores, keys);
  for (int k = 2; k <= NPAD; k <<= 1)
    for (int j = k >> 1; j > 0; j >>= 1)
      rpn_bitonic<<<(BATCH*NPAD)/256, 256, 0, stream>>>(keys, k, j);
  rpn_gather<<<(BATCH*PRE_NMSK + 255)/256, 256, 0, stream>>>(keys, scores, boxes,
                                                             tboxes, keep);
  rpn_nms<<<BATCH, 256, 0, stream>>>(tboxes, keep);
  rpn_output<<<BATCH, 256, 0, stream>>>(tboxes, keep, (float*)d_out,
                                        (float*)d_out + (size_t)BATCH*POST_NMSK*4);
}